// Model_58093727646297
// MI455X (gfx1250) — compile-verified
//
#include <hip/hip_runtime.h>

#define N_NODES 100000
#define FEAT    64
#define MTILES  (N_NODES / 16)   // 6250 exact 16-row tiles

typedef __attribute__((ext_vector_type(2))) float v2f;
typedef __attribute__((ext_vector_type(8))) float v8f;
typedef __attribute__((ext_vector_type(4))) unsigned int v4u;
typedef __attribute__((ext_vector_type(4))) int v4i;
typedef __attribute__((ext_vector_type(8))) int v8i;

// ---------------- degree / norm kernels ----------------
__global__ void k_init_deg(float* deg, int n) {
    int i = blockIdx.x * blockDim.x + threadIdx.x;
    if (i < n) deg[i] = 1.0f;  // self-loop contributes 1 to every node's degree
}

__global__ void k_count_deg(const int* __restrict__ dst, float* deg, int E) {
    int e = blockIdx.x * blockDim.x + threadIdx.x;
    if (e < E) atomicAdd(&deg[dst[e]], 1.0f);
}

__global__ void k_dinv(float* deg, int n) {
    int i = blockIdx.x * blockDim.x + threadIdx.x;
    if (i < n) {
        float d = deg[i];
        deg[i] = (d > 0.0f) ? rsqrtf(d) : 0.0f;  // in-place deg -> dinv
    }
}

// ---------------- fp32 WMMA GEMM: hs = (x @ W^T) * dinv[row] ----------------
// x: [N,K] row-major, W: [64,K] row-major (so B = W^T is K x 64).
// Each wave computes one 16(M) x 64(N) tile via V_WMMA_F32_16X16X4_F32.
// W is staged into LDS by the Tensor Data Mover (one DMA per block) when the
// TDM builtin is available; otherwise by cooperative float4 loads.
__global__ __launch_bounds__(256)
void k_gemm_wmma(const float* __restrict__ x, const float* __restrict__ W,
                 const float* __restrict__ dinv,
                 float* __restrict__ hs, float* __restrict__ agg, int K) {
    __shared__ float sW[FEAT * 256];  // 64 KB max (layer 1); layer 2 uses 16 KB

    const int tid = threadIdx.x;
    const int nW  = FEAT * K;         // 16384 (K=256) or 4096 (K=64) floats

#if __has_builtin(__builtin_amdgcn_tensor_load_to_lds)
    // ---- TDM staging: one tensor_load_to_lds per block, issued by wave 0 ----
    if (tid < 32) {
        unsigned long long wa = (unsigned long long)(const void*)W;
        const unsigned rowElems = (unsigned)(nW / 4);   // 4096 (L1) / 1024 (L2), fits tile_dim0:16b
        // D# group 0: count=1 | lds_addr=0 (sW at LDS offset 0) | global_addr | type=2
        v4u g0;
        g0[0] = 1u;                                  // count=1, is_restore=0, gather off
        g0[1] = 0u;                                  // lds_addr = 0
        g0[2] = (unsigned)(wa & 0xFFFFFFFFu);        // global_addr[31:0]
        g0[3] = (unsigned)((wa >> 32) & 0x01FFFFFFu) // global_addr[56:32]
                | 0x80000000u;                       // type = 2 at bits[127:126]
        // D# group 1: data_size=4B; 2-D tile = 4 rows x rowElems, stride rowElems
        v8i g1;
        g1[0] = 0x00020000;                          // data_size=2 (4 bytes); wg_mask=0
        g1[1] = (int)((rowElems & 0xFFFFu) << 16);   // tensor_dim0[15:0] @ bits 63:48
        g1[2] = (int)(((rowElems >> 16) & 0xFFFFu))  // tensor_dim0[31:16] @ bits 79:64
                | (4 << 16);                         // tensor_dim1 = 4 rows @ bits 111:80
        g1[3] = (int)(rowElems << 16);               // tile_dim0 = rowElems @ bits 127:112
        g1[4] = 4;                                   // tile_dim1 = 4 @ bits 143:128
        g1[5] = (int)rowElems;                       // tensor_dim0_stride[31:0] @ bits 207:160
        g1[6] = 0;
        g1[7] = 0;
        v4i gz4 = {0, 0, 0, 0};                      // groups 2/3 unused (<=2D tensor)
        v8i gz8 = {0, 0, 0, 0, 0, 0, 0, 0};
        __builtin_amdgcn_tensor_load_to_lds(g0, g1, gz4, gz4, gz8, 0);
        __builtin_amdgcn_s_wait_tensorcnt(0);        // wave 0 waits for DMA completion
    }
#else
    // ---- fallback: cooperative vector-load staging ----
    for (int i = tid * 4; i < nW; i += blockDim.x * 4)
        *(float4*)&sW[i] = *(const float4*)&W[i];
#endif
    __syncthreads();
    // The TDM writes LDS through a descriptor the compiler cannot see; escape
    // sW's address with a memory clobber so the B-fragment ds_loads below are
    // not folded away as reads of "never-written" shared memory.
    asm volatile("" :: "v"(&sW[0]) : "memory");

    const int wave = tid >> 5, lane = tid & 31;
    const int tile = blockIdx.x * 8 + wave;
    if (tile >= MTILES) return;              // whole-wave uniform exit: EXEC stays all-1s

    const int half = lane >> 4;              // 0: lanes 0-15, 1: lanes 16-31
    const int l15  = lane & 15;
    const int row0 = tile * 16;

    // A-frag (16x4 f32): lane l15 holds row row0+l15; VGPR pair covers K = k+2*half .. +1
    const float* xa = x + (size_t)(row0 + l15) * K + 2 * half;

    v8f c0 = {}, c1 = {}, c2 = {}, c3 = {};
    for (int k = 0; k < K; k += 4) {
        v2f a = *(const v2f*)(xa + k);
        // B-frag (4x16 f32): column n = nt*16 + l15 of B == row n of W; same K striping as A
        const float* wb = &sW[(size_t)l15 * K + k + 2 * half];
        v2f b0 = *(const v2f*)(wb + 0 * 16 * K);
        v2f b1 = *(const v2f*)(wb + 1 * 16 * K);
        v2f b2 = *(const v2f*)(wb + 2 * 16 * K);
        v2f b3 = *(const v2f*)(wb + 3 * 16 * K);
        c0 = __builtin_amdgcn_wmma_f32_16x16x4_f32(false, a, false, b0, (short)0, c0, false, false);
        c1 = __builtin_amdgcn_wmma_f32_16x16x4_f32(false, a, false, b1, (short)0, c1, false, false);
        c2 = __builtin_amdgcn_wmma_f32_16x16x4_f32(false, a, false, b2, (short)0, c2, false, false);
        c3 = __builtin_amdgcn_wmma_f32_16x16x4_f32(false, a, false, b3, (short)0, c3, false, false);
    }

    // C/D layout: VGPR i holds M = row0 + i + 8*half, N = l15 (per 16x16 f32 table)
#pragma unroll
    for (int i = 0; i < 8; ++i) {
        int row   = row0 + i + 8 * half;
        float s   = dinv[row];
        size_t bo = (size_t)row * FEAT + l15;
        float v;
        v = c0[i] * s; hs[bo +  0] = v; agg[bo +  0] = v;
        v = c1[i] * s; hs[bo + 16] = v; agg[bo + 16] = v;
        v = c2[i] * s; hs[bo + 32] = v; agg[bo + 32] = v;
        v = c3[i] * s; hs[bo + 48] = v; agg[bo + 48] = v;
    }
}

// ---------------- edge aggregation: agg[dst] += hs[src] ----------------
__global__ void k_scatter(const int* __restrict__ src, const int* __restrict__ dst,
                          const float* __restrict__ hs, float* __restrict__ agg, int E) {
    long t = (long)blockIdx.x * blockDim.x + threadIdx.x;
    int e = (int)(t >> 6);
    int f = (int)(t & 63);
    if (e < E) {
        int s = src[e], d = dst[e];
        atomicAdd(&agg[(size_t)d * FEAT + f], hs[(size_t)s * FEAT + f]);
    }
}

// ---------------- epilogue: out = [relu](agg * dinv[node] + bias) ----------------
__global__ void k_post(const float* __restrict__ agg, const float* __restrict__ dinv,
                       const float* __restrict__ bias, float* __restrict__ out,
                       int n, int do_relu) {
    long t = (long)blockIdx.x * blockDim.x + threadIdx.x;
    int node = (int)(t >> 6);
    int f    = (int)(t & 63);
    if (node < n) {
        float v = agg[t] * dinv[node] + bias[f];
        out[t] = do_relu ? fmaxf(v, 0.0f) : v;
    }
}

// ---------------- host launch ----------------
extern "C" void kernel_launch(void* const* d_in, const int* in_sizes, int n_in,
                              void* d_out, int out_size, void* d_ws, size_t ws_size,
                              hipStream_t stream) {
    const float* x1  = (const float*)d_in[0];
    const int*   ei1 = (const int*)d_in[1];
    const float* x2  = (const float*)d_in[2];
    const int*   ei2 = (const int*)d_in[3];
    const float* W0  = (const float*)d_in[4];
    const float* b0  = (const float*)d_in[5];
    const float* W1  = (const float*)d_in[6];
    const float* b1  = (const float*)d_in[7];
    const int E      = in_sizes[1] / 2;
    float* out       = (float*)d_out;

    const size_t nf = (size_t)N_NODES * FEAT;
    char* ws = (char*)d_ws;
    float* dinv = (float*)ws;
    float* bufX = (float*)(ws + (((size_t)N_NODES * 4 + 255) / 256) * 256);
    float* bufY = bufX + nf;
    float* bufZ = bufY + nf;

    const unsigned gN  = (N_NODES + 255) / 256;
    const unsigned gE  = (E + 255) / 256;
    const unsigned gEF = (unsigned)(((long)E * FEAT + 255) / 256);
    const unsigned gNF = (unsigned)((nf + 255) / 256);
    const unsigned gG  = (MTILES + 7) / 8;   // 8 waves (16-row tiles) per 256-thread block

    for (int g = 0; g < 2; ++g) {
        const float* x   = g ? x2 : x1;
        const int*   src = g ? ei2 : ei1;
        const int*   dst = src + E;
        float*       go  = out + (size_t)g * nf;

        k_init_deg <<<gN, 256, 0, stream>>>(dinv, N_NODES);
        k_count_deg<<<gE, 256, 0, stream>>>(dst, dinv, E);
        k_dinv     <<<gN, 256, 0, stream>>>(dinv, N_NODES);

        // layer 1: h = relu(Aggr(x @ W0^T) + b0)
        k_gemm_wmma<<<gG, 256, 0, stream>>>(x, W0, dinv, bufX, bufY, 256);
        k_scatter  <<<gEF, 256, 0, stream>>>(src, dst, bufX, bufY, E);
        k_post     <<<gNF, 256, 0, stream>>>(bufY, dinv, b0, bufX, N_NODES, 1);

        // layer 2: out = Aggr(h @ W1^T) + b1
        k_gemm_wmma<<<gG, 256, 0, stream>>>(bufX, W1, dinv, bufY, bufZ, 64);
        k_scatter  <<<gEF, 256, 0, stream>>>(src, dst, bufY, bufZ, E);
        k_post     <<<gNF, 256, 0, stream>>>(bufZ, dinv, b1, go, N_NODES, 0);
    }
}